// Connection_g_17076789969603
// MI455X (gfx1250) — compile-verified
//
#include <hip/hip_runtime.h>
#include <math.h>

typedef __attribute__((ext_vector_type(2))) float v2f;
typedef __attribute__((ext_vector_type(8))) float v8f;

#define WPB 2              // waves (=samples) per block
#define JSTRIDE 20         // padded row stride for J (80B: 16B-aligned, conflict-free)
#define SLICE 5712         // floats of LDS per wave (multiple of 4)

__global__ __launch_bounds__(WPB * 32) void conn_kernel(
    const float* __restrict__ inp,   // N x 32  [x | v]
    const float* __restrict__ W1,    // 16x16
    const float* __restrict__ b1,    // 16
    const float* __restrict__ W2,    // 256x16
    const float* __restrict__ b2,    // 256
    float* __restrict__ out,         // N x 32  [v | dv]
    int N)
{
    __shared__ float smem[WPB * SLICE];
    const int lane = threadIdx.x & 31;
    const int wid  = threadIdx.x >> 5;
    const int n    = blockIdx.x * WPB + wid;
    const int nn   = (n < N) ? n : 0;          // clamp (no divergence at WMMA)
    const bool valid = (n < N);

    float* Jw   = smem + wid * SLICE;          // 256*20 = 5120 : dA rows, then J in-place
    float* aug  = Jw + 5120;                   // 16*17 = 272   : [g | s] augmented
    float* Atmp = aug + 272;                   // 256           : A = W2 h + b2
    float* xv   = Atmp + 256;                  // 32            : [x | v]
    float* hb   = xv + 32;                     // 16            : h
    float* db   = hb + 16;                     // 16            : 1 - h^2

    // ---- load input row ----
    xv[lane] = inp[nn * 32 + lane];
    __syncthreads();

    // ---- h = tanh(W1 x + b1), d = 1 - h^2 ----
    if (lane < 16) {
        float acc = b1[lane];
        #pragma unroll
        for (int l = 0; l < 16; ++l) acc += W1[lane * 16 + l] * xv[l];
        float hv = tanhf(acc);
        hb[lane] = hv;
        db[lane] = 1.0f - hv * hv;
    }
    __syncthreads();

    // ---- B fragments of M = diag(1-h^2) * W1  (B is 4x16 per chunk) ----
    const int c  = lane & 15;   // N-column (and A-matrix row)
    const int hi = lane >> 4;   // half selector
    v2f bfrag[4];
    #pragma unroll
    for (int kc = 0; kc < 4; ++kc) {
        int k = kc * 4 + hi * 2;
        v2f bf;
        bf.x = db[k]     * W1[k * 16 + c];
        bf.y = db[k + 1] * W1[(k + 1) * 16 + c];
        bfrag[kc] = bf;
    }

    // ---- dA = W2 @ M via 16 tiles of chained V_WMMA_F32_16X16X4_F32;
    //      plain stores of dA rows (no LDS RMW in the hot loop) ----
    #pragma unroll 2
    for (int t = 0; t < 16; ++t) {
        v8f acc = {};
        #pragma unroll
        for (int kc = 0; kc < 4; ++kc) {
            int k = kc * 4 + hi * 2;
            const float* ap = W2 + (t * 16 + c) * 16 + k;
            v2f a; a.x = ap[0]; a.y = ap[1];
            acc = __builtin_amdgcn_wmma_f32_16x16x4_f32(
                false, a, false, bfrag[kc], (short)0, acc, false, false);
        }
        #pragma unroll
        for (int q = 0; q < 8; ++q) {
            int row = q + hi * 8;          // D: lanes 0-15 -> M=q, 16-31 -> M=q+8
            Jw[(t * 16 + row) * JSTRIDE + c] = acc[q];   // dA[(t,row), c]
        }
    }

    // ---- A = W2 h + b2 (8 rows per lane) ----
    #pragma unroll
    for (int e = 0; e < 8; ++e) {
        int p = lane * 8 + e;
        float a = b2[p];
        #pragma unroll
        for (int m = 0; m < 16; ++m) a += W2[p * 16 + m] * hb[m];
        Atmp[p] = a;
    }
    __syncthreads();

    // ---- symmetrize in place: J[i,j,:] = J[j,i,:] = dA[ij,:] + dA[ji,:]
    //      one lane owns each unordered pair -> no cross-lane hazards ----
    for (int pi = lane; pi < 136; pi += 32) {
        int i = 0, rem = pi;
        while (rem >= 16 - i) { rem -= 16 - i; ++i; }
        int j = i + rem;
        float4* p1 = (float4*)(Jw + (i * 16 + j) * JSTRIDE);
        float4* p2 = (float4*)(Jw + (j * 16 + i) * JSTRIDE);
        float4 a0 = p1[0], a1 = p1[1], a2 = p1[2], a3 = p1[3];
        float4 c0 = p2[0], c1 = p2[1], c2 = p2[2], c3 = p2[3];
        a0.x += c0.x; a0.y += c0.y; a0.z += c0.z; a0.w += c0.w;
        a1.x += c1.x; a1.y += c1.y; a1.z += c1.z; a1.w += c1.w;
        a2.x += c2.x; a2.y += c2.y; a2.z += c2.z; a2.w += c2.w;
        a3.x += c3.x; a3.y += c3.y; a3.z += c3.z; a3.w += c3.w;
        p1[0] = a0; p1[1] = a1; p1[2] = a2; p1[3] = a3;
        p2[0] = a0; p2[1] = a1; p2[2] = a2; p2[3] = a3;
    }

    // ---- g = A + A^T into augmented matrix ----
    if (lane < 16) {
        #pragma unroll
        for (int r = 0; r < 16; ++r)
            aug[r * 17 + lane] = Atmp[r * 16 + lane] + Atmp[lane * 16 + r];
    }
    __syncthreads();

    // ---- s[l] = sum_ij v_i v_j J[i,j,l]^2 J[l,i,j] ----
    const float vj = xv[16 + c];
    float part[16];
    #pragma unroll
    for (int l = 0; l < 16; ++l) part[l] = 0.0f;
    for (int ii = 0; ii < 8; ++ii) {
        int i = hi * 8 + ii;
        float w = xv[16 + i] * vj;
        #pragma unroll
        for (int l = 0; l < 16; ++l) {
            float J1 = Jw[(i * 16 + c) * JSTRIDE + l];   // lane-stride 20 -> conflict-free
            float J2 = Jw[(l * 16 + i) * JSTRIDE + c];   // lane-stride 1  -> conflict-free
            part[l] += w * J1 * J1 * J2;
        }
    }
    #pragma unroll
    for (int l = 0; l < 16; ++l) {
        float p = part[l];
        p += __shfl_xor(p, 16, 32);
        p += __shfl_xor(p,  8, 32);
        p += __shfl_xor(p,  4, 32);
        p += __shfl_xor(p,  2, 32);
        p += __shfl_xor(p,  1, 32);
        part[l] = p;
    }
    if (lane == 0) {
        #pragma unroll
        for (int l = 0; l < 16; ++l) aug[l * 17 + 16] = part[l];
    }
    __syncthreads();

    // ---- solve g y = s : Gauss-Jordan with partial pivoting on [g | s] ----
    volatile float* av = aug;
    for (int k = 0; k < 16; ++k) {
        int piv = k; float pv = fabsf(av[k * 17 + k]);
        for (int r = k + 1; r < 16; ++r) {
            float t2 = fabsf(av[r * 17 + k]);
            if (t2 > pv) { pv = t2; piv = r; }
        }
        __syncthreads();
        if (lane < 17 && piv != k) {
            float t1 = av[k * 17 + lane];
            float t2 = av[piv * 17 + lane];
            av[k * 17 + lane]   = t2;
            av[piv * 17 + lane] = t1;
        }
        __syncthreads();
        float pinv = 1.0f / av[k * 17 + k];
        if (lane < 17) av[k * 17 + lane] = av[k * 17 + lane] * pinv;
        __syncthreads();
        if (lane < 17) {
            float akc = av[k * 17 + lane];
            for (int r = 0; r < 16; ++r) {
                if (r == k) continue;
                float f = av[r * 17 + k];          // loads precede the store, in-order LDS
                av[r * 17 + lane] = av[r * 17 + lane] - f * akc;
            }
        }
        __syncthreads();
    }

    // ---- output row: [v | 0.5 * y] ----
    float o;
    if (lane < 16) o = xv[16 + lane];
    else           o = 0.5f * aug[(lane - 16) * 17 + 16];
    if (valid) out[n * 32 + lane] = o;
}

extern "C" void kernel_launch(void* const* d_in, const int* in_sizes, int n_in,
                              void* d_out, int out_size, void* d_ws, size_t ws_size,
                              hipStream_t stream) {
    const float* inp = (const float*)d_in[1];
    const float* W1  = (const float*)d_in[2];
    const float* b1  = (const float*)d_in[3];
    const float* W2  = (const float*)d_in[4];
    const float* b2  = (const float*)d_in[5];
    float* out = (float*)d_out;
    int N = in_sizes[1] / 32;            // 16384
    int blocks = (N + WPB - 1) / WPB;    // 2 samples per block
    conn_kernel<<<blocks, WPB * 32, 0, stream>>>(inp, W1, b1, W2, b2, out, N);
}